// FNetBlock_10118942949971
// MI455X (gfx1250) — compile-verified
//
#include <hip/hip_runtime.h>
#include <math.h>

// FNet mixing: out = Re(FFT2(x, axes=(1,2))) for real x [B=4, S=8192, D=768] fp32.
//
// All matrix work on V_WMMA_F32_16X16X4_F32; shared operands staged via LDS in
// a fragment-ready pair-interleaved layout so every WMMA fragment is a single
// 8B-aligned ds_load_b64 / global_load_b64 (no operand re-packing movs).
//
//  Stage 1: D-axis dense 768-pt DFT (real->complex GEMM):
//           Zr = x @ cosD, Zi = x @ (-sinD)        [32768x768]x[768x768]
//  S-axis 8192 = 64 x 128 Cooley-Tukey, n = n1 + 64*n2, k = 128*k1 + k2:
//  Stage 2: per (b,n1): T[k2] = sum_n2 Z[n1+64*n2] * W128^(n2 k2),
//           then twiddle * W8192^(n1 k2); written IN PLACE at s = n1 + 64*k2.
//  Stage 3: per (b,k2): out[128*k1+k2] = Tr@cos64 + Ti@sin64 (real part only).

typedef __attribute__((ext_vector_type(2))) float v2f;
typedef __attribute__((ext_vector_type(8))) float v8f;

#define BATCH 4
#define SEQ   8192
#define DIM   768
#define BSROWS (BATCH * SEQ)   // 32768
#define KC    32               // stage-1 K chunk staged through LDS

static __device__ __forceinline__ v8f wmma4(v2f a, v2f b, v8f c) {
  // D = A(16x4, f32) * B(4x16, f32) + C(16x16, f32)
  return __builtin_amdgcn_wmma_f32_16x16x4_f32(
      /*neg_a=*/false, a, /*neg_b=*/false, b,
      /*c_mod=*/(short)0, c, /*reuse_a=*/false, /*reuse_b=*/false);
}

// A-matrix 16x4 f32 fragment (ISA 7.12.2): VGPR0 = K=0 (lanes 0-15) / K=2
// (lanes 16-31); VGPR1 = K=1 / K=3. Row M = lane % 16. K is contiguous in A
// (row-major), so the two K values are one aligned 64-bit load.
static __device__ __forceinline__ v2f load_a2(const float* A, int lda, int row0,
                                              int k0) {
  int lane = threadIdx.x & 31;
  int m = lane & 15;
  int kk = (lane >> 4) << 1;               // 0 or 2
  return *(const v2f*)(A + (size_t)(row0 + m) * lda + k0 + kk);
}

// B-matrix 4x16 f32 fragment from pair-interleaved LDS layout:
//   element (k, c) at offset ((k>>1)*ncols + c)*2 + (k&1)
// -> lane (n = lane%16, kk = 2*(lane/16)) reads one aligned 64-bit pair.
static __device__ __forceinline__ v2f load_b_pair(const float* B, int ncols,
                                                  int k0, int col0) {
  int lane = threadIdx.x & 31;
  int n = lane & 15;
  int kk = (lane >> 4) << 1;
  return *(const v2f*)(B + ((((k0 + kk) >> 1) * ncols + col0 + n) << 1));
}

static __device__ __forceinline__ int pair_idx(int k, int c, int ncols) {
  return (((k >> 1) * ncols + c) << 1) | (k & 1);
}

// C/D 16x16 f32: VGPR r holds M = r (lanes 0-15) / M = 8+r (lanes 16-31), N = lane%16.
static __device__ __forceinline__ void store_c(float* __restrict__ C, int ldc,
                                               int row0, int col0, v8f c) {
  int lane = threadIdx.x & 31;
  int n = lane & 15;
  int half = lane >> 4;
#pragma unroll
  for (int r = 0; r < 8; ++r)
    C[(size_t)(row0 + half * 8 + r) * ldc + col0 + n] = c[r];
}

// ---------------- table init (rebuilt every launch; deterministic) ----------
__global__ void init_tables(float* __restrict__ cosD, float* __restrict__ nsinD,
                            float* __restrict__ c128, float* __restrict__ s128,
                            float* __restrict__ ns128,
                            float* __restrict__ c64, float* __restrict__ s64) {
  const float TWO_PI = 6.283185307179586f;
  int idx = blockIdx.x * blockDim.x + threadIdx.x;
  if (idx < DIM * DIM) {
    int i = idx / DIM, j = idx - i * DIM;
    int m = (i * j) % DIM;                       // exact modular phase
    float s, c; sincosf(TWO_PI * (float)m / (float)DIM, &s, &c);
    cosD[idx] = c; nsinD[idx] = -s;
  } else if (idx < DIM * DIM + 128 * 128) {
    int t = idx - DIM * DIM;
    int i = t >> 7, j = t & 127;
    int m = (i * j) & 127;
    float s, c; sincosf(TWO_PI * (float)m / 128.0f, &s, &c);
    c128[t] = c; s128[t] = s; ns128[t] = -s;
  } else if (idx < DIM * DIM + 128 * 128 + 64 * 64) {
    int t = idx - DIM * DIM - 128 * 128;
    int i = t >> 6, j = t & 63;
    int m = (i * j) & 63;
    float s, c; sincosf(TWO_PI * (float)m / 64.0f, &s, &c);
    c64[t] = c; s64[t] = s;
  }
}

// ---------------- Stage 1: Zr = x@cosD, Zi = x@nsinD ------------------------
// Block: 256 thr (8 waves); 128 rows x 128 cols output tile.
// Per K-chunk (32): stage x rows (128x32, row-major) and cos/nsin
// (2 x 32x128, pair-interleaved) in LDS; coalesced b128 global loads.
__global__ __launch_bounds__(256) void stage1_dft_d(
    const float* __restrict__ x, const float* __restrict__ cosD,
    const float* __restrict__ nsinD, float* __restrict__ Zr,
    float* __restrict__ Zi) {
  __shared__ float lds_a[128][KC];           // x tile, row-major   (16 KB)
  __shared__ float lds_b[2][KC * 128];       // cos/nsin, paired    (32 KB)

  int tid = threadIdx.x;
  int wave = tid >> 5;
  int blockRow = blockIdx.x * 128;           // 256 blocks
  int row0l = wave * 16;                     // local row tile
  int colg = blockIdx.y * 128;               // 6 groups

  v8f z = {};
  v8f accR[8], accI[8];
#pragma unroll
  for (int t = 0; t < 8; ++t) { accR[t] = z; accI[t] = z; }

  for (int kc = 0; kc < DIM; kc += KC) {
    __syncthreads();                         // previous chunk fully consumed
    // stage A: 128 rows x 32 K  (1024 float4, 4 per thread, coalesced)
#pragma unroll
    for (int i = 0; i < 4; ++i) {
      int j = tid + i * 256;
      int row = j >> 3, c4 = (j & 7) << 2;
      float4 v = *(const float4*)(x + (size_t)(blockRow + row) * DIM + kc + c4);
      lds_a[row][c4 + 0] = v.x; lds_a[row][c4 + 1] = v.y;
      lds_a[row][c4 + 2] = v.z; lds_a[row][c4 + 3] = v.w;
    }
    // stage B: 2 matrices x 32 K x 128 cols (2048 float4, 8 per thread),
    // scattered into the pair-interleaved layout.
#pragma unroll
    for (int i = 0; i < 8; ++i) {
      int j = tid + i * 256;
      int mat = j >> 10, rem = j & 1023;
      int krow = rem >> 5, c4 = (rem & 31) << 2;
      const float* src = mat ? nsinD : cosD;
      float4 v = *(const float4*)(src + (size_t)(kc + krow) * DIM + colg + c4);
      float* d = lds_b[mat];
      d[pair_idx(krow, c4 + 0, 128)] = v.x;
      d[pair_idx(krow, c4 + 1, 128)] = v.y;
      d[pair_idx(krow, c4 + 2, 128)] = v.z;
      d[pair_idx(krow, c4 + 3, 128)] = v.w;
    }
    __syncthreads();

    for (int k = 0; k < KC; k += 4) {
      v2f a = load_a2(&lds_a[0][0], KC, row0l, k);
#pragma unroll
      for (int t = 0; t < 8; ++t) {
        v2f br = load_b_pair(lds_b[0], 128, k, t * 16);
        accR[t] = wmma4(a, br, accR[t]);
        v2f bi = load_b_pair(lds_b[1], 128, k, t * 16);
        accI[t] = wmma4(a, bi, accI[t]);
      }
    }
  }
#pragma unroll
  for (int t = 0; t < 8; ++t) {
    store_c(Zr, DIM, blockRow + row0l, colg + t * 16, accR[t]);
    store_c(Zi, DIM, blockRow + row0l, colg + t * 16, accI[t]);
  }
}

// ---------------- Stage 2: radix-128 pass + twiddle, in place ---------------
// Block = (colg, n1, b); 8 waves; wave w owns k2 rows [w*16,+16), 16 cols.
// 128x16 complex Z panel staged once (pair-interleaved); the post-load barrier
// also makes the in-place write-back safe (all global reads complete first).
__global__ __launch_bounds__(256) void stage2_radix128(
    float* __restrict__ Zr, float* __restrict__ Zi,
    const float* __restrict__ c128, const float* __restrict__ s128,
    const float* __restrict__ ns128) {
  __shared__ float ldsr[128 * 16];           // 8 KB, paired
  __shared__ float ldsi[128 * 16];           // 8 KB, paired

  int tid = threadIdx.x;
  int b = blockIdx.z;                        // 4
  int n1 = blockIdx.y;                       // 64
  int col = blockIdx.x * 16;                 // 48 tiles
  int wave = tid >> 5;
  int row0 = wave * 16;                      // k2 tile

  float* zr = Zr + (size_t)b * SEQ * DIM;
  float* zi = Zi + (size_t)b * SEQ * DIM;
  const float* br_base = zr + (size_t)n1 * DIM;    // row n2 -> + n2*64*DIM
  const float* bi_base = zi + (size_t)n1 * DIM;
  const int ldb = 64 * DIM;

  // stage Z panel: 2 x 128 rows x 16 cols (1024 float4, 4 per thread)
#pragma unroll
  for (int i = 0; i < 4; ++i) {
    int j = tid + i * 256;
    int mat = j >> 9, rem = j & 511;
    int krow = rem >> 2, c4 = (rem & 3) << 2;
    const float* g = (mat ? bi_base : br_base) + (size_t)krow * ldb + col + c4;
    float4 v = *(const float4*)g;
    float* d = mat ? ldsi : ldsr;
    d[pair_idx(krow, c4 + 0, 16)] = v.x;
    d[pair_idx(krow, c4 + 1, 16)] = v.y;
    d[pair_idx(krow, c4 + 2, 16)] = v.z;
    d[pair_idx(krow, c4 + 3, 16)] = v.w;
  }
  __syncthreads();

  v8f accR = {}, accI = {};
  for (int k = 0; k < 128; k += 4) {
    v2f aC = load_a2(c128, 128, row0, k);
    v2f aS = load_a2(s128, 128, row0, k);
    v2f aN = load_a2(ns128, 128, row0, k);
    v2f bR = load_b_pair(ldsr, 16, k, 0);
    v2f bI = load_b_pair(ldsi, 16, k, 0);
    accR = wmma4(aC, bR, accR);
    accR = wmma4(aS, bI, accR);
    accI = wmma4(aC, bI, accI);
    accI = wmma4(aN, bR, accI);
  }

  int lane = tid & 31;
  int n = lane & 15;
  int half = lane >> 4;
  const float TWO_PI = 6.283185307179586f;
#pragma unroll
  for (int r = 0; r < 8; ++r) {
    int k2 = row0 + half * 8 + r;
    int m = (n1 * k2) & (SEQ - 1);
    float sphi, cphi;
    sincosf(TWO_PI * (float)m / (float)SEQ, &sphi, &cphi);
    float tr = accR[r], ti = accI[r];
    float trp = tr * cphi + ti * sphi;       // T * e^{-i phi}
    float tip = ti * cphi - tr * sphi;
    size_t off = (size_t)(n1 + 64 * k2) * DIM + col + n;
    zr[off] = trp;
    zi[off] = tip;
  }
}

// ---------------- Stage 3: radix-64 pass, real part -> d_out ----------------
// Block = (colg, k2, b); 4 waves; wave w owns k1 rows [w*16,+16), 16 cols.
// Input rows 64*k2 + n1 are contiguous; 64x16 complex panel staged (paired).
__global__ __launch_bounds__(128) void stage3_radix64(
    const float* __restrict__ Zr, const float* __restrict__ Zi,
    const float* __restrict__ c64, const float* __restrict__ s64,
    float* __restrict__ out) {
  __shared__ float ldsr[64 * 16];            // 4 KB, paired
  __shared__ float ldsi[64 * 16];            // 4 KB, paired

  int tid = threadIdx.x;
  int b = blockIdx.z;                        // 4
  int k2 = blockIdx.y;                       // 128
  int col = blockIdx.x * 16;                 // 48 tiles
  int wave = tid >> 5;
  int row0 = wave * 16;                      // k1 tile

  const float* tr = Zr + (size_t)b * SEQ * DIM + (size_t)(64 * k2) * DIM;
  const float* ti = Zi + (size_t)b * SEQ * DIM + (size_t)(64 * k2) * DIM;

  // stage panel: 2 x 64 rows x 16 cols (512 float4, 4 per thread)
#pragma unroll
  for (int i = 0; i < 4; ++i) {
    int j = tid + i * 128;
    int mat = j >> 8, rem = j & 255;
    int krow = rem >> 2, c4 = (rem & 3) << 2;
    const float* g = (mat ? ti : tr) + (size_t)krow * DIM + col + c4;
    float4 v = *(const float4*)g;
    float* d = mat ? ldsi : ldsr;
    d[pair_idx(krow, c4 + 0, 16)] = v.x;
    d[pair_idx(krow, c4 + 1, 16)] = v.y;
    d[pair_idx(krow, c4 + 2, 16)] = v.z;
    d[pair_idx(krow, c4 + 3, 16)] = v.w;
  }
  __syncthreads();

  v8f acc = {};
  for (int k = 0; k < 64; k += 4) {
    v2f aC = load_a2(c64, 64, row0, k);
    v2f aS = load_a2(s64, 64, row0, k);
    v2f bR = load_b_pair(ldsr, 16, k, 0);
    v2f bI = load_b_pair(ldsi, 16, k, 0);
    acc = wmma4(aC, bR, acc);
    acc = wmma4(aS, bI, acc);
  }

  int lane = tid & 31;
  int n = lane & 15;
  int half = lane >> 4;
  float* o = out + (size_t)b * SEQ * DIM;
#pragma unroll
  for (int r = 0; r < 8; ++r) {
    int k1 = row0 + half * 8 + r;
    o[(size_t)(128 * k1 + k2) * DIM + col + n] = acc[r];
  }
}

extern "C" void kernel_launch(void* const* d_in, const int* in_sizes, int n_in,
                              void* d_out, int out_size, void* d_ws, size_t ws_size,
                              hipStream_t stream) {
  (void)in_sizes; (void)n_in; (void)out_size; (void)ws_size;
  const float* x = (const float*)d_in[0];
  float* out = (float*)d_out;
  float* ws = (float*)d_ws;

  const size_t BSD = (size_t)BSROWS * DIM;        // 25,165,824 floats (96 MB)
  float* Zr    = ws;
  float* Zi    = Zr + BSD;
  float* cosD  = Zi + BSD;
  float* nsinD = cosD + (size_t)DIM * DIM;
  float* c128  = nsinD + (size_t)DIM * DIM;
  float* s128  = c128 + 128 * 128;
  float* ns128 = s128 + 128 * 128;
  float* c64   = ns128 + 128 * 128;
  float* s64   = c64 + 64 * 64;
  // total ws use: 2*BSD + 2*768^2 + 3*128^2 + 2*64^2 floats (~197 MB)

  int tab_n = DIM * DIM + 128 * 128 + 64 * 64;    // 610,304
  init_tables<<<(tab_n + 255) / 256, 256, 0, stream>>>(cosD, nsinD, c128, s128,
                                                       ns128, c64, s64);
  stage1_dft_d<<<dim3(BSROWS / 128, DIM / 128, 1), 256, 0, stream>>>(
      x, cosD, nsinD, Zr, Zi);
  stage2_radix128<<<dim3(DIM / 16, 64, BATCH), 256, 0, stream>>>(
      Zr, Zi, c128, s128, ns128);
  stage3_radix64<<<dim3(DIM / 16, 128, BATCH), 128, 0, stream>>>(
      Zr, Zi, c64, s64, out);
}